// Net_74474732913335
// MI455X (gfx1250) — compile-verified
//
#include <hip/hip_runtime.h>

// ============================================================================
// MI455X (gfx1250) implementation of the 3-branch GCN reference.
//
// Roofline: edge scatter/gather dominates (~9-10 GB/fwd -> ~0.45ms @23.3TB/s);
// dense math is ~20 GFLOP, so we keep fp32 and use V_WMMA_F32_16X16X4_F32 for
// every linear layer. GEMM stages both A and B tiles in LDS so the inner loop
// is branch-free: ds_load_b64 + ds_load_2addr_b32 + v_wmma only.
//
// Param flattening assumption: jax pytree order (dict keys sorted at every
// level). Top-level inputs keep setup_inputs() insertion order (indices 0..8),
// params leaves follow at index 9+ in sorted-key order:
//   colconv(b,w) conv1(b,w) conv2(b,w) conv3(b,w) finallin(b,w)
//   i1: bn1(b,g,m,v) bn2(b,g,m,v) c1(b,w) c2(b,w) c3(b,w)
//   i2: bn1 bn2 bn3 c1 c2 c3    lin1(b,w) lin2(b,w)
//   mlp[0..2](b,w)  mlp2[0..2](b,w)
// ============================================================================

#define CDIV(a, b) (((a) + (b) - 1) / (b))
#define GG 64

typedef float v2f __attribute__((ext_vector_type(2)));
typedef float v8f __attribute__((ext_vector_type(8)));

// ---------------------------------------------------------------------------
// fp32 WMMA GEMM:  C[M,Nout] = epi(A[M,K] @ B[K,Nout])
// One wave = one 16x16 output tile (V_WMMA_F32_16X16X4_F32), 4 waves/block.
// B tile (Kpad x 16) staged in LDS by the whole block; each wave stages its
// own A tile (16 x Kpad, row stride Kpad+4 to dodge bank-stride conflicts).
// Zero-padded K rows in the B tile annihilate A padding, so the wmma loop is
// completely unguarded.
// MODE bits: 1=+bias, 2=relu, 4=batchnorm(g,b,m,v), 8=relu-after-bn
// RELUA: relu applied to A elements during staging (fuses relu(prev layer)).
// ---------------------------------------------------------------------------
template <int MODE, bool RELUA>
__global__ void gemm_wmma_f32(
    const float* __restrict__ A, int lda,
    const float* __restrict__ B, int ldb,
    float* __restrict__ C, int ldc,
    int M, int K, int Nout,
    const float* __restrict__ bias,
    const float* __restrict__ bng, const float* __restrict__ bnb,
    const float* __restrict__ bnm, const float* __restrict__ bnv)
{
  extern __shared__ float smem[];            // [Kpad*16] B ++ [4][16][Kpad+4] A
  const int lane = threadIdx.x & 31;
  const int wave = threadIdx.x >> 5;
  const int half = lane >> 4;                // 0: lanes 0-15, 1: lanes 16-31
  const int mn   = lane & 15;
  const int Kpad = (K + 3) & ~3;
  const int ldsa = Kpad + 4;
  const int col0 = blockIdx.y * 16;

  float* Bs = smem;
  float* As = smem + Kpad * 16 + wave * 16 * ldsa;   // per-wave A tile

  // ---- stage B tile (cols col0..col0+15), zero-padded ----
  for (int idx = threadIdx.x; idx < Kpad * 16; idx += blockDim.x) {
    int k = idx >> 4, n = idx & 15;
    int gn = col0 + n;
    Bs[idx] = (k < K && gn < Nout) ? B[(size_t)k * ldb + gn] : 0.0f;
  }
  __syncthreads();

  const int row0 = (blockIdx.x * 4 + wave) * 16;
  if (row0 >= M) return;                     // wave-uniform exit (after barrier)

  // ---- stage A tile (16 rows x Kpad), coalesced along K, zero-padded ----
  for (int m = 0; m < 16; ++m) {
    int row = row0 + m; if (row > M - 1) row = M - 1;   // clamp; stores masked
    const float* Ar = A + (size_t)row * lda;
    for (int k = lane; k < Kpad; k += 32) {
      float v = (k < K) ? Ar[k] : 0.0f;
      if (RELUA) v = fmaxf(v, 0.0f);
      As[m * ldsa + k] = v;
    }
  }

  // ---- branch-free wmma loop: pure LDS reads ----
  const float* Aln = As + mn * ldsa;         // this lane's A row (m = mn)
  v8f acc = {0.f, 0.f, 0.f, 0.f, 0.f, 0.f, 0.f, 0.f};
  for (int k0 = 0; k0 < Kpad; k0 += 4) {
    int ka = k0 + 2 * half;                  // A frag: lane holds K = 2*half+v
    v2f a; a.x = Aln[ka];           a.y = Aln[ka + 1];
    v2f b; b.x = Bs[ka * 16 + mn];  b.y = Bs[ka * 16 + 16 + mn];
    acc = __builtin_amdgcn_wmma_f32_16x16x4_f32(
        false, a, false, b, (short)0, acc, false, false);
  }

  // ---- epilogue ----
  const int gn = col0 + mn;
  float bv = 0.0f, sc = 1.0f, sh = 0.0f;
  if (gn < Nout) {
    if (MODE & 1) bv = bias[gn];
    if (MODE & 4) {
      float inv = rsqrtf(bnv[gn] + 1e-5f);
      sc = bng[gn] * inv;
      sh = bnb[gn] - bnm[gn] * sc;
    }
  }
  #pragma unroll
  for (int g = 0; g < 8; ++g) {              // C frag: vgpr g -> row g + 8*half
    int row = row0 + g + 8 * half;
    if (row < M && gn < Nout) {
      float v = acc[g];
      if (MODE & 1) v += bv;
      if (MODE & 2) v = fmaxf(v, 0.0f);
      if (MODE & 4) v = v * sc + sh;
      if (MODE & 8) v = fmaxf(v, 0.0f);
      C[(size_t)row * ldc + gn] = v;
    }
  }
}

// ---------------------------------------------------------------------------
// Elementwise / GCN helper kernels
// ---------------------------------------------------------------------------
__global__ void fill_kernel(float* p, float v, int n) {
  int i = blockIdx.x * blockDim.x + threadIdx.x;
  if (i < n) p[i] = v;
}

__global__ void fill2d_kernel(float* p, int ldp, int rows, int cols, float v) {
  int i = blockIdx.x * blockDim.x + threadIdx.x;
  if (i < rows * cols) p[(size_t)(i / cols) * ldp + (i % cols)] = v;
}

__global__ void deg_kernel(const int* __restrict__ dst, int E, float* __restrict__ deg) {
  int i = blockIdx.x * blockDim.x + threadIdx.x;
  if (i < E) unsafeAtomicAdd(&deg[dst[i]], 1.0f);
}

__global__ void rsqrt_kernel(float* p, int n) {
  int i = blockIdx.x * blockDim.x + threadIdx.x;
  if (i < n) p[i] = rsqrtf(p[i]);            // deg >= 1 (self loop), always valid
}

// OUT[i][c] = bias[c] + H[i][c] * dinv[i]^2   (self-loop message + bias)
__global__ void gcn_self_kernel(const float* __restrict__ H,
                                const float* __restrict__ dinv,
                                const float* __restrict__ bias,
                                float* __restrict__ OUT, int n, int C) {
  int idx = blockIdx.x * blockDim.x + threadIdx.x;
  if (idx >= n * C) return;
  int i = idx / C, c = idx - i * C;
  float d = dinv[i];
  OUT[idx] = bias[c] + H[idx] * d * d;
}

// OUT[dst] += H[src] * dinv[src]*dinv[dst]; one thread = (edge, 4 channels)
__global__ void gcn_edge_kernel(const float* __restrict__ H,
                                const float* __restrict__ dinv,
                                const int* __restrict__ src,
                                const int* __restrict__ dst,
                                int E, float* __restrict__ OUT, int C4) {
  int idx = blockIdx.x * blockDim.x + threadIdx.x;
  if (idx >= E * C4) return;
  int e  = idx / C4;
  int c  = (idx - e * C4) * 4;
  int s  = src[e], d = dst[e];
  float w = dinv[s] * dinv[d];
  const int C = C4 * 4;
  float4 h = *reinterpret_cast<const float4*>(H + (size_t)s * C + c);
  float* o = OUT + (size_t)d * C + c;
  unsafeAtomicAdd(o + 0, h.x * w);
  unsafeAtomicAdd(o + 1, h.y * w);
  unsafeAtomicAdd(o + 2, h.z * w);
  unsafeAtomicAdd(o + 3, h.w * w);
}

// duplicate inception1 x2 (cols 16..31 -> 32..47 of the [N,80] buffer)
__global__ void copycols_kernel(float* __restrict__ I1, int n) {
  int idx = blockIdx.x * blockDim.x + threadIdx.x;
  if (idx >= n * 16) return;
  int i = idx >> 4, c = idx & 15;
  I1[(size_t)i * 80 + 32 + c] = I1[(size_t)i * 80 + 16 + c];
}

// ---------------------------------------------------------------------------
// Mean pool (segment mean over batch -> [G, C] slice with stride ldp)
// ---------------------------------------------------------------------------
__global__ void pool_accum_kernel(const float* __restrict__ X,
                                  const int* __restrict__ batch,
                                  int n, int C, float* __restrict__ P, int ldp) {
  int idx = blockIdx.x * blockDim.x + threadIdx.x;
  if (idx >= n * C) return;
  int i = idx / C, c = idx - i * C;
  unsafeAtomicAdd(&P[(size_t)batch[i] * ldp + c], X[idx]);
}

__global__ void pool_count_kernel(const int* __restrict__ batch, int n,
                                  float* __restrict__ cnt) {
  int i = blockIdx.x * blockDim.x + threadIdx.x;
  if (i < n) unsafeAtomicAdd(&cnt[batch[i]], 1.0f);
}

__global__ void pool_div_kernel(float* __restrict__ P, int ldp, int rows, int C,
                                const float* __restrict__ cnt) {
  int idx = blockIdx.x * blockDim.x + threadIdx.x;
  if (idx >= rows * C) return;
  int g = idx / C, c = idx - g * C;
  P[(size_t)g * ldp + c] /= fmaxf(cnt[g], 1.0f);
}

// ---------------------------------------------------------------------------
// finallin([xout|xcolout]) -> log_softmax, one thread per graph
// ---------------------------------------------------------------------------
__global__ void final_kernel(const float* __restrict__ O6a,
                             const float* __restrict__ O6b,
                             const float* __restrict__ w,   // [12,5]
                             const float* __restrict__ b,   // [5]
                             float* __restrict__ out) {     // [G,5]
  int g = blockIdx.x * blockDim.x + threadIdx.x;
  if (g >= GG) return;
  float in[12];
  #pragma unroll
  for (int i = 0; i < 6; ++i) { in[i] = O6a[g * 6 + i]; in[6 + i] = O6b[g * 6 + i]; }
  float o[5];
  #pragma unroll
  for (int j = 0; j < 5; ++j) {
    float acc = b[j];
    #pragma unroll
    for (int i = 0; i < 12; ++i) acc += in[i] * w[i * 5 + j];
    o[j] = acc;
  }
  float mx = o[0];
  #pragma unroll
  for (int j = 1; j < 5; ++j) mx = fmaxf(mx, o[j]);
  float s = 0.0f;
  #pragma unroll
  for (int j = 0; j < 5; ++j) s += expf(o[j] - mx);
  float lse = mx + logf(s);
  #pragma unroll
  for (int j = 0; j < 5; ++j) out[g * 5 + j] = o[j] - lse;
}

// ---------------------------------------------------------------------------
// Host orchestration
// ---------------------------------------------------------------------------
static void launch_gemm(hipStream_t s,
                        const float* A, int lda, const float* B, int ldb,
                        float* C, int ldc, int M, int K, int Nout,
                        const float* bias,
                        const float* g, const float* bb, const float* m, const float* v,
                        int mode, int reluA) {
  int Kpad = (K + 3) & ~3;
  dim3 grid(CDIV(M, 64), CDIV(Nout, 16));
  size_t lds = ((size_t)Kpad * 16 + (size_t)4 * 16 * (Kpad + 4)) * sizeof(float);
#define GL(MD, RA) gemm_wmma_f32<MD, RA><<<grid, 128, lds, s>>>( \
    A, lda, B, ldb, C, ldc, M, K, Nout, bias, g, bb, m, v)
  if (reluA) {
    GL(0, true);
  } else {
    switch (mode) {
      case 1:  GL(1, false);  break;
      case 3:  GL(3, false);  break;
      case 7:  GL(7, false);  break;
      case 15: GL(15, false); break;
      default: GL(0, false);  break;
    }
  }
#undef GL
}

extern "C" void kernel_launch(void* const* d_in, const int* in_sizes, int n_in,
                              void* d_out, int out_size, void* d_ws, size_t ws_size,
                              hipStream_t stream) {
  const float* x16     = (const float*)d_in[0];
  const float* x20     = (const float*)d_in[1];
  const float* xcol    = (const float*)d_in[2];
  const int*   edge16  = (const int*)d_in[3];
  const int*   edge20  = (const int*)d_in[4];
  const int*   edgecol = (const int*)d_in[5];
  const int*   batch16 = (const int*)d_in[6];
  const int*   batch20 = (const int*)d_in[7];
  const int*   batchcol= (const int*)d_in[8];
  const int N = in_sizes[0] / 33;
  const int E = in_sizes[3] / 2;

  auto F = [&](int i) { return (const float*)d_in[i]; };
  // params leaves (sorted-key pytree order; see header comment)
  const float *colconv_b = F(9),  *colconv_w = F(10);
  const float *conv1_b = F(11),   *conv1_w = F(12);
  const float *conv2_b = F(13),   *conv2_w = F(14);
  const float *conv3_b = F(15),   *conv3_w = F(16);
  const float *fin_b = F(17),     *fin_w = F(18);
  const float *i1bn1_b = F(19), *i1bn1_g = F(20), *i1bn1_m = F(21), *i1bn1_v = F(22);
  const float *i1bn2_b = F(23), *i1bn2_g = F(24), *i1bn2_m = F(25), *i1bn2_v = F(26);
  const float *i1c1_b = F(27), *i1c1_w = F(28);
  const float *i1c2_b = F(29), *i1c2_w = F(30);
  const float *i1c3_b = F(31), *i1c3_w = F(32);
  const float *i2bn1_b = F(33), *i2bn1_g = F(34), *i2bn1_m = F(35), *i2bn1_v = F(36);
  const float *i2bn2_b = F(37), *i2bn2_g = F(38), *i2bn2_m = F(39), *i2bn2_v = F(40);
  const float *i2bn3_b = F(41), *i2bn3_g = F(42), *i2bn3_m = F(43), *i2bn3_v = F(44);
  const float *i2c1_b = F(45), *i2c1_w = F(46);
  const float *i2c2_b = F(47), *i2c2_w = F(48);
  const float *i2c3_b = F(49), *i2c3_w = F(50);
  const float *lin1_b = F(51), *lin1_w = F(52);
  const float *lin2_b = F(53), *lin2_w = F(54);
  const float *mlp0_b = F(55), *mlp0_w = F(56);
  const float *mlp1_b = F(57), *mlp1_w = F(58);
  const float *mlp2_b = F(59), *mlp2_w = F(60);
  const float *m2l0_b = F(61), *m2l0_w = F(62);
  const float *m2l1_b = F(63), *m2l1_w = F(64);
  const float *m2l2_b = F(65), *m2l2_w = F(66);

  // ---- workspace layout (~192 MB) ----
  float* ws = (float*)d_ws;
  size_t off = 0;
  auto alloc = [&](size_t n) { float* r = ws + off; off += n; return r; };
  float* bufA  = alloc((size_t)N * 128);   // GEMM output (h = x @ W)
  float* bufB  = alloc((size_t)N * 128);   // GCN scatter output
  float* bufI1 = alloc((size_t)N * 80);    // inception1 concat
  float* bufI2 = alloc((size_t)N * 140);   // inception2 concat
  float* dinv  = alloc((size_t)N);         // deg -> rsqrt(deg)
  float* pool2 = alloc((size_t)GG * 280);  // [h16 | h20]
  float* poolc = alloc((size_t)GG * 140);  // hcol
  float* cnt   = alloc((size_t)GG);
  float* T1    = alloc((size_t)GG * 128);
  float* T2    = alloc((size_t)GG * 64);
  float* T3    = alloc((size_t)GG * 32);
  float* O6a   = alloc((size_t)GG * 6);
  float* O6b   = alloc((size_t)GG * 6);
  (void)ws_size; (void)n_in; (void)out_size;

  const int TB = 256;

  auto gcn_scatter = [&](const float* H, const float* bias, float* OUT,
                         const int* src, const int* dst, int C) {
    gcn_self_kernel<<<CDIV(N * C, TB), TB, 0, stream>>>(H, dinv, bias, OUT, N, C);
    int C4 = C / 4;
    gcn_edge_kernel<<<CDIV(E * C4, TB), TB, 0, stream>>>(H, dinv, src, dst, E, OUT, C4);
  };

  auto run_branch = [&](const float* X, int inDim, const float* w1, const float* b1,
                        const int* edge, const int* batch, float* Pdst, int ldp) {
    const int* src = edge;
    const int* dst = edge + E;
    // symmetric GCN normalization: deg = 1 (self loop) + indegree; dinv = rsqrt
    fill_kernel<<<CDIV(N, TB), TB, 0, stream>>>(dinv, 1.0f, N);
    deg_kernel<<<CDIV(E, TB), TB, 0, stream>>>(dst, E, dinv);
    rsqrt_kernel<<<CDIV(N, TB), TB, 0, stream>>>(dinv, N);
    // conv1: h = X @ W1 ; out = scatter + b1 ; relu fused into next GEMM A-load
    launch_gemm(stream, X, inDim, w1, 128, bufA, 128, N, inDim, 128,
                nullptr, nullptr, nullptr, nullptr, nullptr, 0, 0);
    gcn_scatter(bufA, b1, bufB, src, dst, 128);
    // conv2 (A = relu(conv1 out))
    launch_gemm(stream, bufB, 128, conv2_w, 64, bufA, 64, N, 128, 64,
                nullptr, nullptr, nullptr, nullptr, nullptr, 0, 1);
    gcn_scatter(bufA, conv2_b, bufB, src, dst, 64);
    // conv3 (A = relu(conv2 out)); no relu after conv3
    launch_gemm(stream, bufB, 64, conv3_w, 32, bufA, 32, N, 64, 32,
                nullptr, nullptr, nullptr, nullptr, nullptr, 0, 1);
    gcn_scatter(bufA, conv3_b, bufB, src, dst, 32);
    // inception1 -> bufI1[N,80] = [x1 | x2 | x2 | x5]
    launch_gemm(stream, bufB, 32, i1c1_w, 16, bufI1 + 0, 80, N, 32, 16,
                i1c1_b, i1bn2_g, i1bn2_b, i1bn2_m, i1bn2_v, 7, 0);   // x1=bn2(relu(.))
    launch_gemm(stream, bufI1, 80, i1c2_w, 16, bufI1 + 16, 80, N, 16, 16,
                i1c2_b, i1bn2_g, i1bn2_b, i1bn2_m, i1bn2_v, 7, 0);   // x2
    copycols_kernel<<<CDIV(N * 16, TB), TB, 0, stream>>>(bufI1, N);  // x3 = x2
    launch_gemm(stream, bufI1, 80, i1c3_w, 32, bufI1 + 48, 80, N, 16, 32,
                i1c3_b, i1bn1_g, i1bn1_b, i1bn1_m, i1bn1_v, 7, 0);   // x5 (from x1)
    // inception2 -> bufI2[N,140], each branch relu(bn(relu(lin)))
    launch_gemm(stream, bufI1, 80, i2c1_w, 80, bufI2 + 0, 140, N, 80, 80,
                i2c1_b, i2bn1_g, i2bn1_b, i2bn1_m, i2bn1_v, 15, 0);
    launch_gemm(stream, bufI1, 80, i2c2_w, 40, bufI2 + 80, 140, N, 80, 40,
                i2c2_b, i2bn2_g, i2bn2_b, i2bn2_m, i2bn2_v, 15, 0);
    launch_gemm(stream, bufI1, 80, i2c3_w, 20, bufI2 + 120, 140, N, 80, 20,
                i2c3_b, i2bn3_g, i2bn3_b, i2bn3_m, i2bn3_v, 15, 0);
    // mean pool -> Pdst[G,140] (stride ldp)
    fill2d_kernel<<<CDIV(GG * 140, TB), TB, 0, stream>>>(Pdst, ldp, GG, 140, 0.0f);
    fill_kernel<<<1, GG, 0, stream>>>(cnt, 0.0f, GG);
    pool_accum_kernel<<<CDIV(N * 140, TB), TB, 0, stream>>>(bufI2, batch, N, 140, Pdst, ldp);
    pool_count_kernel<<<CDIV(N, TB), TB, 0, stream>>>(batch, N, cnt);
    pool_div_kernel<<<CDIV(GG * 140, TB), TB, 0, stream>>>(Pdst, ldp, GG, 140, cnt);
  };

  run_branch(x16,  33, conv1_w,   conv1_b,   edge16,  batch16,  pool2,       280);
  run_branch(x20,  33, conv1_w,   conv1_b,   edge20,  batch20,  pool2 + 140, 280);
  run_branch(xcol, 39, colconv_w, colconv_b, edgecol, batchcol, poolc,       140);

  // head 1: [h16|h20] -> lin1 -> mlp -> O6a
  launch_gemm(stream, pool2, 280, lin1_w, 128, T1, 128, GG, 280, 128,
              lin1_b, nullptr, nullptr, nullptr, nullptr, 3, 0);
  launch_gemm(stream, T1, 128, mlp0_w, 64, T2, 64, GG, 128, 64,
              mlp0_b, nullptr, nullptr, nullptr, nullptr, 3, 0);
  launch_gemm(stream, T2, 64, mlp1_w, 32, T3, 32, GG, 64, 32,
              mlp1_b, nullptr, nullptr, nullptr, nullptr, 3, 0);
  launch_gemm(stream, T3, 32, mlp2_w, 6, O6a, 6, GG, 32, 6,
              mlp2_b, nullptr, nullptr, nullptr, nullptr, 1, 0);

  // head 2: hcol -> lin2 -> mlp2 -> O6b
  launch_gemm(stream, poolc, 140, lin2_w, 128, T1, 128, GG, 140, 128,
              lin2_b, nullptr, nullptr, nullptr, nullptr, 3, 0);
  launch_gemm(stream, T1, 128, m2l0_w, 64, T2, 64, GG, 128, 64,
              m2l0_b, nullptr, nullptr, nullptr, nullptr, 3, 0);
  launch_gemm(stream, T2, 64, m2l1_w, 32, T3, 32, GG, 64, 32,
              m2l1_b, nullptr, nullptr, nullptr, nullptr, 3, 0);
  launch_gemm(stream, T3, 32, m2l2_w, 6, O6b, 6, GG, 32, 6,
              m2l2_b, nullptr, nullptr, nullptr, nullptr, 1, 0);

  // finallin + log_softmax
  final_kernel<<<1, GG, 0, stream>>>(O6a, O6b, fin_w, fin_b, (float*)d_out);
}